// QuantumLayer_29222957482684
// MI455X (gfx1250) — compile-verified
//
#include <hip/hip_runtime.h>

typedef __attribute__((ext_vector_type(2))) float v2f;
typedef __attribute__((ext_vector_type(8))) float v8f;

#define PI_CONST 3.1415f
#define TPW 4  // 16-sample tiles per wave

// ---------------------------------------------------------------------------
// Kernel 1: build Mall[q][row][col] = M_q (4 real symmetric 16x16 matrices)
// from the weights. One block, runs once, writes 4 KB to d_ws.
// Uses precise libm trig: this result is shared by all 1M samples.
// ---------------------------------------------------------------------------
__global__ __launch_bounds__(256) void build_M_kernel(const float* __restrict__ w,
                                                      float* __restrict__ Mall) {
  __shared__ float Ur[256];  // Ur[k*16 + j] : row k, col j of U (real part)
  __shared__ float Ui[256];
  const int tid = threadIdx.x;

  if (tid < 16) {
    // Thread j applies the weight-layer circuit to basis state e_j -> column j of U.
    float sr[16], si[16];
#pragma unroll
    for (int k = 0; k < 16; ++k) { sr[k] = (k == tid) ? 1.f : 0.f; si[k] = 0.f; }

    for (int layer = 0; layer < 2; ++layer) {
      for (int q = 0; q < 4; ++q) {
        const int mask = 1 << (3 - q);   // axis q is bit (3-q), MSB-first
        float a, c, s;
        // RX(a) = [[c, -i s], [-i s, c]]
        a = w[layer * 12 + q * 3 + 0]; c = cosf(0.5f * a); s = sinf(0.5f * a);
        for (int b = 0; b < 16; ++b) {
          if (b & mask) continue;
          const int b1 = b | mask;
          float r0 = sr[b], i0 = si[b], r1 = sr[b1], i1 = si[b1];
          sr[b]  = c * r0 + s * i1;  si[b]  = c * i0 - s * r1;
          sr[b1] = c * r1 + s * i0;  si[b1] = c * i1 - s * r0;
        }
        // RY(a) = [[c, -s], [s, c]]
        a = w[layer * 12 + q * 3 + 1]; c = cosf(0.5f * a); s = sinf(0.5f * a);
        for (int b = 0; b < 16; ++b) {
          if (b & mask) continue;
          const int b1 = b | mask;
          float r0 = sr[b], i0 = si[b], r1 = sr[b1], i1 = si[b1];
          sr[b]  = c * r0 - s * r1;  si[b]  = c * i0 - s * i1;
          sr[b1] = s * r0 + c * r1;  si[b1] = s * i0 + c * i1;
        }
        // RZ(a) = diag(e^{-ia/2}, e^{+ia/2})
        a = w[layer * 12 + q * 3 + 2]; c = cosf(0.5f * a); s = sinf(0.5f * a);
        for (int b = 0; b < 16; ++b) {
          if (b & mask) continue;
          const int b1 = b | mask;
          float r0 = sr[b], i0 = si[b], r1 = sr[b1], i1 = si[b1];
          sr[b]  = c * r0 + s * i0;  si[b]  = c * i0 - s * r0;
          sr[b1] = c * r1 - s * i1;  si[b1] = c * i1 + s * r1;
        }
      }
      // CNOT chain (c=q, t=q+1): for control bit 1, swap target 0<->1
      for (int q = 0; q < 3; ++q) {
        const int cm = 1 << (3 - q), tm = 1 << (2 - q);
        for (int b = 0; b < 16; ++b) {
          if ((b & cm) && !(b & tm)) {
            const int b1 = b | tm;
            float tr = sr[b]; sr[b] = sr[b1]; sr[b1] = tr;
            float ti = si[b]; si[b] = si[b1]; si[b1] = ti;
          }
        }
      }
    }
#pragma unroll
    for (int k = 0; k < 16; ++k) { Ur[k * 16 + tid] = sr[k]; Ui[k * 16 + tid] = si[k]; }
  }
  __syncthreads();

  // M_q[row][col] = sum_k sign_q(k) * (Ur[k,row]Ur[k,col] + Ui[k,row]Ui[k,col])
  // 1024 entries / 256 threads = 4 each; flat index e == q*256 + row*16 + col.
#pragma unroll
  for (int u = 0; u < 4; ++u) {
    const int e = tid * 4 + u;
    const int q = e >> 8, rem = e & 255, row = rem >> 4, col = rem & 15;
    float acc = 0.f;
    for (int k = 0; k < 16; ++k) {
      const float sign = ((k >> (3 - q)) & 1) ? -1.f : 1.f;
      acc += sign * (Ur[k * 16 + row] * Ur[k * 16 + col] +
                     Ui[k * 16 + row] * Ui[k * 16 + col]);
    }
    Mall[e] = acc;
  }
}

// ---------------------------------------------------------------------------
// Fast device math. tanh via the exact identity 1 - 2/(e^{2x}+1)
// (v_exp_f32 + v_rcp_f32; correct limits at +-inf). Trig arguments are
// bounded in (-pi/2, pi/2) by construction, so native v_sin/v_cos
// (no range reduction) are safe.
// ---------------------------------------------------------------------------
__device__ __forceinline__ float fast_tanh(float x) {
  float e = __expf(2.f * x);                       // v_exp_f32 based
  return 1.f - 2.f * __builtin_amdgcn_rcpf(e + 1.f);  // v_rcp_f32
}

// ---------------------------------------------------------------------------
// Kernel 2: per wave, 16 samples per tile, TPW tiles per wave.
// ev_q = v^T M_q v via V_WMMA_F32_16X16X4_F32 (K accumulated in 4 chunks).
// Samples ride the N (column) dimension. All per-lane fragment values are
// built from the factorization v[b] = t01[b>>2]*t23[b&3], so the half-wave
// selector `hi` appears in exactly 4 cndmasks — no dynamic register indexing.
// ---------------------------------------------------------------------------
__global__ __launch_bounds__(256) void qcircuit_wmma_kernel(
    const float* __restrict__ x, const float* __restrict__ Mall,
    float* __restrict__ out, int B, int ntiles) {
  const int lane = threadIdx.x & 31;
  const int lo = lane & 15;
  const int hi = lane >> 4;  // 0: lanes 0-15, 1: lanes 16-31
  const int wave = (blockIdx.x * blockDim.x + threadIdx.x) >> 5;

  // A-matrix fragments (16x4 f32 layout): lane l, VGPR j holds
  // A[M = l%16][K = j + 2*(l>=16)]. A_k = rows 4k..4k+3 of symmetric M_q.
  // `hi` only enters the *address* computation here (fine).
  float afrag[4][4][2];
#pragma unroll
  for (int q = 0; q < 4; ++q)
#pragma unroll
    for (int k = 0; k < 4; ++k)
#pragma unroll
      for (int j = 0; j < 2; ++j)
        afrag[q][k][j] = Mall[q * 256 + (4 * k + j + 2 * hi) * 16 + lo];

  for (int t = 0; t < TPW; ++t) {
    const int tile = wave * TPW + t;   // wave-uniform
    if (tile >= ntiles) return;        // uniform exit: EXEC stays all-ones for WMMA

    const int s = tile * 16 + lo;      // this lane's sample (both halves mirror it)
    const int sc = (s < B) ? s : (B - 1);
    const float4 xv = *reinterpret_cast<const float4*>(x + (size_t)sc * 8);

    // Encoded product state factors: v[b] = t01[b>>2] * t23[b&3]
    // h_q = (PI/2) * tanh(x_q)  in (-pi/2, pi/2)  -> native trig is safe
    const float h0 = 0.5f * PI_CONST * fast_tanh(xv.x);
    const float h1 = 0.5f * PI_CONST * fast_tanh(xv.y);
    const float h2 = 0.5f * PI_CONST * fast_tanh(xv.z);
    const float h3 = 0.5f * PI_CONST * fast_tanh(xv.w);
    const float c0 = __cosf(h0), s0 = __sinf(h0);
    const float c1 = __cosf(h1), s1 = __sinf(h1);
    const float c2 = __cosf(h2), s2 = __sinf(h2);
    const float c3 = __cosf(h3), s3 = __sinf(h3);
    const float a00 = c0 * c1, a01 = c0 * s1, a10 = s0 * c1, a11 = s0 * s1; // t01[0..3]
    const float b00 = c2 * c3, b01 = c2 * s3, b10 = s2 * c3, b11 = s2 * s3; // t23[0..3]

    // Half-wave selections: the ONLY places `hi` touches data (4 cndmasks).
    const float u0 = hi ? b10 : b00;  // t23[0 + 2*hi]
    const float u1 = hi ? b11 : b01;  // t23[1 + 2*hi]
    const float w0 = hi ? a10 : a00;  // t01[0 + 2*hi]
    const float w1 = hi ? a11 : a01;  // t01[1 + 2*hi]

    // B fragments (4x16 f32 layout): lane l, VGPR j holds B_k[j + 2*hi][l%16]
    //   = v[4k + j + 2*hi] = t01[k] * t23[j + 2*hi]
    const v2f bf0 = {a00 * u0, a00 * u1};
    const v2f bf1 = {a01 * u0, a01 * u1};
    const v2f bf2 = {a10 * u0, a10 * u1};
    const v2f bf3 = {a11 * u0, a11 * u1};

    // v^T fragment in C/D layout: vt[r] = v[r + 8*hi] = t01[(r>>2)+2*hi] * t23[r&3]
    const float vt[8] = {w0 * b00, w0 * b01, w0 * b10, w0 * b11,
                         w1 * b00, w1 * b01, w1 * b10, w1 * b11};

    float ev[4];
#pragma unroll
    for (int q = 0; q < 4; ++q) {
      v8f acc = {};
      {
        v2f A0 = {afrag[q][0][0], afrag[q][0][1]};
        acc = __builtin_amdgcn_wmma_f32_16x16x4_f32(false, A0, false, bf0,
                                                    (short)0, acc, false, false);
        v2f A1 = {afrag[q][1][0], afrag[q][1][1]};
        acc = __builtin_amdgcn_wmma_f32_16x16x4_f32(false, A1, false, bf1,
                                                    (short)0, acc, false, false);
        v2f A2 = {afrag[q][2][0], afrag[q][2][1]};
        acc = __builtin_amdgcn_wmma_f32_16x16x4_f32(false, A2, false, bf2,
                                                    (short)0, acc, false, false);
        v2f A3 = {afrag[q][3][0], afrag[q][3][1]};
        acc = __builtin_amdgcn_wmma_f32_16x16x4_f32(false, A3, false, bf3,
                                                    (short)0, acc, false, false);
      }
      // ev_q[i] = sum_m v_i[m] * D[m][i]; D lane l reg r = D[r + 8*hi][l%16]
      float p = 0.f;
#pragma unroll
      for (int r = 0; r < 8; ++r) p += vt[r] * acc[r];
      p += __shfl_xor(p, 16, 32);  // add the two row-halves
      ev[q] = p;
    }

    if (hi == 0 && s < B) {
      float4 o = {ev[0], ev[1], ev[2], ev[3]};
      *reinterpret_cast<float4*>(out + (size_t)s * 4) = o;  // out is (B,4) row-major
    }
  }
}

// ---------------------------------------------------------------------------
extern "C" void kernel_launch(void* const* d_in, const int* in_sizes, int n_in,
                              void* d_out, int out_size, void* d_ws, size_t ws_size,
                              hipStream_t stream) {
  const float* x = (const float*)d_in[0];       // (B, 8) fp32
  const float* w = (const float*)d_in[1];       // (2, 4, 3) fp32
  float* out = (float*)d_out;                   // (B, 4) fp32
  float* Mall = (float*)d_ws;                   // 4 * 16 * 16 fp32 = 4 KB

  const int B = in_sizes[0] / 8;

  build_M_kernel<<<1, 256, 0, stream>>>(w, Mall);

  const int ntiles = (B + 15) / 16;
  const int wavesNeeded = (ntiles + TPW - 1) / TPW;
  const int nblocks = (wavesNeeded + 7) / 8;    // 8 waves (256 threads) per block
  qcircuit_wmma_kernel<<<nblocks, 256, 0, stream>>>(x, Mall, out, B, ntiles);
}